// VQ_15539191677467
// MI455X (gfx1250) — compile-verified
//
#include <hip/hip_runtime.h>

#ifndef __has_builtin
#define __has_builtin(x) 0
#endif

typedef float v2f __attribute__((ext_vector_type(2)));
typedef float v8f __attribute__((ext_vector_type(8)));
typedef int   v4i_ __attribute__((vector_size(16)));
typedef __attribute__((address_space(1))) v4i_* as1_v4i;
typedef __attribute__((address_space(3))) v4i_* as3_v4i;
typedef __attribute__((address_space(1))) int*  as1_i;
typedef __attribute__((address_space(3))) int*  as3_i;

#define BB   8
#define NIN  256
#define NN   8192
#define DD   64
#define KK   512
#define NT   128

// ---- dynamic LDS layout (float indices) ----
// z tile / ze live pair-interleaved: element (k,n) -> (k>>1)*STRIDE + n*2 + (k&1)
#define ZSP_STRIDE 260                       // 64 pair-rows of 256 floats (+pad)
#define ZEP_STRIDE 260                       // 32 pair-rows of 256 floats (+pad)
#define WS_STRIDE  260                       // W 64 x 256 (padded, row-major)
#define ZE_STRIDE  132                       // phase-4 zq gather tile (row-major)
#define EMB_STRIDE 68                        // emb block 64 x 64 (padded, row-major)
#define WS_OFF   16896                       // >= 64*ZSP_STRIDE = 16640
#define EMB0_OFF (WS_OFF)                    // emb double-buffer 0 (aliases W)
#define EMB1_OFF (WS_OFF + 64 * EMB_STRIDE)  // emb double-buffer 1
#define SQ_OFF   (WS_OFF + 2 * 64 * EMB_STRIDE) // emb_sq[512]
#define SV_OFF   (SQ_OFF + 512)              // per-rowgroup min values 4*128
#define SI_OFF   (SV_OFF + 512)              // per-rowgroup min indices 4*128
#define IND_OFF  (SI_OFF + 512)              // final ind[128]
#define SMEM_FLOATS (WS_OFF + 64 * WS_STRIDE)   // 33536 floats
#define SMEM_BYTES  (SMEM_FLOATS * 4)           // 134144 bytes

// ---- CDNA5 async global->LDS copies, ASYNCcnt-tracked ----
__device__ __forceinline__ void g2l_b128(const float* g, float* l) {
#if __has_builtin(__builtin_amdgcn_global_load_async_to_lds_b128)
    __builtin_amdgcn_global_load_async_to_lds_b128(
        (as1_v4i)(v4i_*)(void*)g,
        (as3_v4i)(v4i_*)(void*)l, 0, 0);
#else
    unsigned int loff = (unsigned int)(unsigned long long)l;
    asm volatile("global_load_async_to_lds_b128 %0, %1, off"
                 :: "v"(loff), "v"(g) : "memory");
#endif
}

__device__ __forceinline__ void g2l_b32(const float* g, float* l) {
#if __has_builtin(__builtin_amdgcn_global_load_async_to_lds_b32)
    __builtin_amdgcn_global_load_async_to_lds_b32(
        (as1_i)(int*)(void*)g,
        (as3_i)(int*)(void*)l, 0, 0);
#else
    unsigned int loff = (unsigned int)(unsigned long long)l;
    asm volatile("global_load_async_to_lds_b32 %0, %1, off"
                 :: "v"(loff), "v"(g) : "memory");
#endif
}

__device__ __forceinline__ void wait_async0() {
#if __has_builtin(__builtin_amdgcn_s_wait_asynccnt)
    __builtin_amdgcn_s_wait_asynccnt(0);
#else
    asm volatile("s_wait_asynccnt 0x0" ::: "memory");
#endif
}

__global__ void vq_hist_init(const float* __restrict__ ih, float* __restrict__ hist) {
    int i = blockIdx.x * blockDim.x + threadIdx.x;
    if (i < KK) hist[i] = ih[i];
}

__global__ void __launch_bounds__(256, 2)
vq_fused(const float* __restrict__ z, const float* __restrict__ W,
         const float* __restrict__ emb, float* __restrict__ zq,
         float* __restrict__ hist)
{
    extern __shared__ float smem[];
    const int tid = threadIdx.x;
    const int b   = blockIdx.y;
    const int n0  = blockIdx.x * NT;

    const int ln  = tid & 31;
    const int wv  = tid >> 5;
    const int rg  = wv & 3;          // row-group of 16 (4 groups)
    const int cg  = wv >> 2;         // column-group of 64 (2 groups)
    const int m0  = rg * 16;
    const int c0  = cg * 64;
    const int lx  = ln & 15;
    const int lhi = (ln >> 4) & 1;   // lane >= 16 half
    const int klo = lhi * 2;         // K sub-offset for A/B fragments

    const float* zb = z + (size_t)b * NIN * NN;

    // ---- async-stage W (64 x 256, row-major) into LDS ----
    #pragma unroll
    for (int it = 0; it < 16; ++it) {
        int e = it * 256 + tid;          // 4096 float4
        int row = e >> 6;
        int c4  = e & 63;
        g2l_b128(W + (size_t)row * NIN + c4 * 4,
                 smem + WS_OFF + row * WS_STRIDE + c4 * 4);
    }

    const v8f vzero = {0.f,0.f,0.f,0.f,0.f,0.f,0.f,0.f};
    v8f acc[4] = {vzero, vzero, vzero, vzero};

    // ---- phase 1: ze(64 x 128) = W @ z-tile via V_WMMA_F32_16X16X4_F32 ----
    // z tile staged pair-interleaved with b32 async copies: B frag = 1 ds_load_b64
    const int zk = tid >> 7;             // per-thread fixed (k parity group, n)
    const int znc = tid & 127;
    for (int kc = 0; kc < 2; ++kc) {
        if (kc) __syncthreads();         // all waves done reading previous chunk
        #pragma unroll
        for (int it = 0; it < 64; ++it) {
            int k = it * 2 + zk;         // 0..127
            g2l_b32(zb + (size_t)(kc * 128 + k) * NN + n0 + znc,
                    smem + (k >> 1) * ZSP_STRIDE + znc * 2 + (k & 1));
        }
        wait_async0();
        __syncthreads();
        #pragma unroll
        for (int k = 0; k < 128; k += 4) {
            const int ka = kc * 128 + k;
            v2f a = *(const v2f*)(smem + WS_OFF + (m0 + lx) * WS_STRIDE + ka + klo);
            #pragma unroll
            for (int t = 0; t < 4; ++t) {
                const int col = c0 + t * 16 + lx;
                v2f bb = *(const v2f*)(smem + ((k + klo) >> 1) * ZSP_STRIDE + col * 2);
                acc[t] = __builtin_amdgcn_wmma_f32_16x16x4_f32(
                    false, a, false, bb, (short)0, acc[t], false, false);
            }
        }
    }
    __syncthreads();

    // ---- phase 1.5: prefetch emb block 0 (async), spill ze pair-interleaved,
    //      compute emb_sq ----
    #pragma unroll
    for (int it = 0; it < 4; ++it) {     // emb block kb=0 -> buffer 0
        int e = it * 256 + tid;          // 1024 float4 = 64 rows x 16 f4
        int row = e >> 4;
        int c4  = e & 15;
        g2l_b128(emb + (size_t)row * DD + c4 * 4,
                 smem + EMB0_OFF + row * EMB_STRIDE + c4 * 4);
    }

    #pragma unroll
    for (int t = 0; t < 4; ++t)
        #pragma unroll
        for (int v = 0; v < 8; ++v) {
            int row = m0 + v + lhi * 8;
            smem[(row >> 1) * ZEP_STRIDE + (c0 + t * 16 + lx) * 2 + (row & 1)] = acc[t][v];
        }

    for (int r = tid; r < KK; r += 256) {
        const float4* ep = (const float4*)(emb + (size_t)r * DD);
        float s = 0.f;
        #pragma unroll
        for (int j = 0; j < 16; ++j) {
            float4 q = ep[j];
            s += q.x * q.x + q.y * q.y + q.z * q.z + q.w * q.w;
        }
        smem[SQ_OFF + r] = s;
    }
    wait_async0();
    __syncthreads();

    // ---- phase 2: cross = emb @ ze, fused argmin, double-buffered staging ----
    float minv[4] = {3.4e38f, 3.4e38f, 3.4e38f, 3.4e38f};
    int   mini[4] = {0, 0, 0, 0};

    for (int kb = 0; kb < 8; ++kb) {
        const float* cur = smem + ((kb & 1) ? EMB1_OFF : EMB0_OFF);
        if (kb < 7) {                    // async-prefetch next block into other buffer
            float* nxt = smem + (((kb + 1) & 1) ? EMB1_OFF : EMB0_OFF);
            #pragma unroll
            for (int it = 0; it < 4; ++it) {
                int e = it * 256 + tid;
                int row = e >> 4;
                int c4  = e & 15;
                g2l_b128(emb + (size_t)((kb + 1) * 64 + row) * DD + c4 * 4,
                         nxt + row * EMB_STRIDE + c4 * 4);
            }
        }

        v8f cacc[4] = {vzero, vzero, vzero, vzero};
        #pragma unroll
        for (int k = 0; k < 64; k += 4) {
            v2f a = *(const v2f*)(cur + (m0 + lx) * EMB_STRIDE + k + klo);
            #pragma unroll
            for (int t = 0; t < 4; ++t) {
                const int col = c0 + t * 16 + lx;
                v2f bb = *(const v2f*)(smem + ((k + klo) >> 1) * ZEP_STRIDE + col * 2);
                cacc[t] = __builtin_amdgcn_wmma_f32_16x16x4_f32(
                    false, a, false, bb, (short)0, cacc[t], false, false);
            }
        }
        #pragma unroll
        for (int t = 0; t < 4; ++t)
            #pragma unroll
            for (int v = 0; v < 8; ++v) {
                int code = kb * 64 + m0 + v + lhi * 8;
                float val = smem[SQ_OFF + code] - 2.0f * cacc[t][v];
                if (val < minv[t]) { minv[t] = val; mini[t] = code; }
            }

        wait_async0();                   // our prefetch for kb+1 has landed
        __syncthreads();                 // publish; all waves done with 'cur'
    }

    // ---- phase 3: per-column argmin reduction (lane pair, then 4 row-groups) ----
    #pragma unroll
    for (int t = 0; t < 4; ++t) {
        float ov = __shfl_xor(minv[t], 16, 32);
        int   oi = __shfl_xor(mini[t], 16, 32);
        if (ov < minv[t]) { minv[t] = ov; mini[t] = oi; }
    }
    if (ln < 16) {
        #pragma unroll
        for (int t = 0; t < 4; ++t) {
            int c = c0 + t * 16 + lx;
            smem[SV_OFF + rg * 128 + c] = minv[t];
            ((int*)smem)[SI_OFF + rg * 128 + c] = mini[t];
        }
    }
    __syncthreads();
    if (tid < 128) {
        int c = tid;
        float bv = smem[SV_OFF + c];
        int   bi = ((int*)smem)[SI_OFF + c];
        #pragma unroll
        for (int g = 1; g < 4; ++g) {
            float v = smem[SV_OFF + g * 128 + c];
            if (v < bv) { bv = v; bi = ((int*)smem)[SI_OFF + g * 128 + c]; }
        }
        ((int*)smem)[IND_OFF + c] = bi;
        atomicAdd(&hist[bi], 1.0f);      // exact integer-valued f32 adds: deterministic
    }
    __syncthreads();

    // ---- phase 4: gather emb[ind] transposed into LDS, store zq as float4 ----
    {
        int c = tid & 127;
        int h = tid >> 7;                // d halves 0..31 / 32..63
        int code = ((int*)smem)[IND_OFF + c];
        const float4* er = (const float4*)(emb + (size_t)code * DD + h * 32);
        #pragma unroll
        for (int j = 0; j < 8; ++j) {
            float4 q = er[j];
            int d = h * 32 + j * 4;
            smem[(d + 0) * ZE_STRIDE + c] = q.x;
            smem[(d + 1) * ZE_STRIDE + c] = q.y;
            smem[(d + 2) * ZE_STRIDE + c] = q.z;
            smem[(d + 3) * ZE_STRIDE + c] = q.w;
        }
    }
    __syncthreads();
    {
        float* outb = zq + (size_t)b * DD * NN + n0;
        #pragma unroll
        for (int i = 0; i < 8; ++i) {
            int e = i * 256 + tid;       // 2048 float4 = 64 rows x 32 f4
            int row = e >> 5;
            int c4  = e & 31;
            float4 q = *(const float4*)(smem + row * ZE_STRIDE + c4 * 4);
            *(float4*)(outb + (size_t)row * NN + c4 * 4) = q;
        }
    }
}

extern "C" void kernel_launch(void* const* d_in, const int* in_sizes, int n_in,
                              void* d_out, int out_size, void* d_ws, size_t ws_size,
                              hipStream_t stream) {
    (void)in_sizes; (void)n_in; (void)out_size; (void)d_ws; (void)ws_size;
    const float* z   = (const float*)d_in[0];
    const float* W   = (const float*)d_in[1];
    const float* emb = (const float*)d_in[2];
    const float* ih  = (const float*)d_in[3];
    float* zq   = (float*)d_out;
    float* hist = zq + (size_t)BB * DD * NN;

    (void)hipFuncSetAttribute((const void*)vq_fused,
                              hipFuncAttributeMaxDynamicSharedMemorySize, SMEM_BYTES);

    vq_hist_init<<<2, 256, 0, stream>>>(ih, hist);
    vq_fused<<<dim3(NN / NT, BB), 256, SMEM_BYTES, stream>>>(z, W, emb, zq, hist);
}